// APoTQuantizer_6940667150461
// MI455X (gfx1250) — compile-verified
//
#include <hip/hip_runtime.h>
#include <stdint.h>

// APoT quantizer, eval forward — bandwidth-bound elementwise kernel.
// Strategy: stream x as float4 (global_load_b128 / global_store_b128, NT hints),
// stage the 243-entry codebook into LDS with gfx1250 async global->LDS copies,
// then do an exact branchless searchsorted (top 3 levels in registers, bottom 5
// in LDS) matching the reference's tie-break semantics bit-for-bit.

typedef float v4f __attribute__((ext_vector_type(4)));

#define BLOCK_DIM 256
#define K_CB      243   // codebook entries for (8,2) APoT
#define CB_PAD    256   // padded to power of two with +INF

__device__ __forceinline__ float apot_q1(float xv, float a,
                                         float c127, float c63, float c191,
                                         float c31, float c95, float c159, float c223,
                                         const float* s_cb) {
  // IEEE division to match the reference exactly (quantization decisions are
  // threshold compares; a 1-ulp reciprocal shortcut could flip bins).
  float xn = xv / a;
  xn = fminf(1.0f, fmaxf(-1.0f, xn));

  // j = #{k : cb[k] < xn} == jnp.searchsorted(cb, xn, side='left')
  // branchless scan over the 256-padded array: if (cb[j+s-1] < xn) j += s
  // levels s=128,64,32 use register-cached pivots (no LDS traffic).
  int j = (c127 < xn) ? 128 : 0;
  float t2 = (j != 0) ? c191 : c63;
  if (t2 < xn) j += 64;
  float lo = (j & 64) ? c95  : c31;
  float hi = (j & 64) ? c223 : c159;
  float t3 = (j & 128) ? hi : lo;
  if (t3 < xn) j += 32;
  if (s_cb[j + 15] < xn) j += 16;
  if (s_cb[j +  7] < xn) j += 8;
  if (s_cb[j +  3] < xn) j += 4;
  if (s_cb[j +  1] < xn) j += 2;
  if (s_cb[j     ] < xn) j += 1;

  // i = clip(j, 1, K-1); left/right neighbors; tie -> left (smaller index)
  int i1 = (j < 1) ? 1 : ((j > K_CB - 1) ? (K_CB - 1) : j);
  float l = s_cb[i1 - 1];
  float r = s_cb[i1];
  float q = (fabsf(xn - l) <= fabsf(xn - r)) ? l : r;
  return q * a;
}

__global__ __launch_bounds__(BLOCK_DIM) void apot_quant_kernel(
    const float* __restrict__ x, const float* __restrict__ alpha,
    const float* __restrict__ cb, float* __restrict__ out, long long n) {
  __shared__ float s_cb[CB_PAD];
  const int t = threadIdx.x;

  // Pad tail with +INF so the 256-wide branchless search never over-counts.
  if (t >= K_CB) s_cb[t] = __builtin_inff();

  // gfx1250 async copy: codebook global -> LDS without a VGPR round trip.
  // Tracked by ASYNCcnt; one b32 per lane for t < 243.
  if (t < K_CB) {
    uint32_t lds_off = (uint32_t)(uintptr_t)&s_cb[t];      // low 32 bits = LDS byte offset
    uint64_t gaddr   = (uint64_t)(uintptr_t)(cb + t);
    asm volatile("global_load_async_to_lds_b32 %0, %1, off"
                 :: "v"(lds_off), "v"(gaddr)
                 : "memory");
  }
  asm volatile("s_wait_asynccnt 0" ::: "memory");
  __syncthreads();

  const float a = fabsf(alpha[0]) + 1e-5f;

  // Register-cache the top 3 search levels (uniform values).
  const float c127 = s_cb[127];
  const float c63  = s_cb[63],  c191 = s_cb[191];
  const float c31  = s_cb[31],  c95  = s_cb[95];
  const float c159 = s_cb[159], c223 = s_cb[223];

  const long long nvec   = n >> 2;
  const v4f* __restrict__ x4 = (const v4f*)x;
  v4f* __restrict__ o4       = (v4f*)out;
  const long long stride = (long long)gridDim.x * BLOCK_DIM;

  for (long long i = (long long)blockIdx.x * BLOCK_DIM + t; i < nvec; i += stride) {
    if (i + stride < nvec)
      __builtin_prefetch(&x4[i + stride], 0, 3);           // global_prefetch_b8

    v4f v = __builtin_nontemporal_load(&x4[i]);            // global_load_b128 (NT)
    v4f r;
    r.x = apot_q1(v.x, a, c127, c63, c191, c31, c95, c159, c223, s_cb);
    r.y = apot_q1(v.y, a, c127, c63, c191, c31, c95, c159, c223, s_cb);
    r.z = apot_q1(v.z, a, c127, c63, c191, c31, c95, c159, c223, s_cb);
    r.w = apot_q1(v.w, a, c127, c63, c191, c31, c95, c159, c223, s_cb);
    __builtin_nontemporal_store(r, &o4[i]);                // global_store_b128 (NT)
  }

  // Scalar tail (n % 4) — n is divisible by 4 for this shape, but stay general.
  const long long base = nvec << 2;
  const long long tail = n - base;
  if (blockIdx.x == 0 && t < tail) {
    out[base + t] = apot_q1(x[base + t], a, c127, c63, c191, c31, c95, c159, c223, s_cb);
  }
}

extern "C" void kernel_launch(void* const* d_in, const int* in_sizes, int n_in,
                              void* d_out, int out_size, void* d_ws, size_t ws_size,
                              hipStream_t stream) {
  const float* x     = (const float*)d_in[0];   // (4096, 8192) fp32
  const float* alpha = (const float*)d_in[1];   // scalar
  const float* cb    = (const float*)d_in[2];   // (243,) sorted fp32
  float* out = (float*)d_out;

  const long long n    = (long long)in_sizes[0];
  long long nvec   = n >> 2;
  long long blocks = (nvec + BLOCK_DIM - 1) / BLOCK_DIM;
  if (blocks < 1) blocks = 1;
  if (blocks > 4096) blocks = 4096;             // persistent blocks, grid-stride

  apot_quant_kernel<<<dim3((unsigned)blocks), dim3(BLOCK_DIM), 0, stream>>>(
      x, alpha, cb, out, n);
}